// PinSAGE_44908178047437
// MI455X (gfx1250) — compile-verified
//
#include <hip/hip_runtime.h>
#include <hip/hip_bf16.h>

// ---------------------------------------------------------------------------
// PinSAGE: per-node mean of first-10 out-neighbors (edge order), fallback to
// self feature, then 128x128 linear.  f32 WMMA (v_wmma_f32_16x16x4_f32) for
// the linear layer; gather-bound overall, so fp32 matrix math is "free".
// B fragments (W^T) are hoisted into registers once per block.
// ---------------------------------------------------------------------------

#define D_FEAT   128
#define KNBR     10
#define CAP      64          // per-node bucket capacity (>= realistic max degree)
#define WT_LD    132         // padded LDS stride for W^T (bank-conflict dodge)
#define AGG_LD   132         // padded LDS stride for agg tile
#define NKSTEP   (D_FEAT / 4)   // 32 WMMA k-steps

typedef float v2f __attribute__((ext_vector_type(2)));
typedef float v8f __attribute__((ext_vector_type(8)));

// ---- kernel 1: zero per-node edge counters --------------------------------
__global__ void ps_zero_counts(int* __restrict__ cnt, int n) {
    int i = blockIdx.x * blockDim.x + threadIdx.x;
    if (i < n) cnt[i] = 0;
}

// ---- kernel 2: scatter edge ids into per-node buckets ---------------------
__global__ void ps_fill_buckets(const int* __restrict__ row,
                                int* __restrict__ cnt,
                                int* __restrict__ bucket,
                                int E) {
    int e = blockIdx.x * blockDim.x + threadIdx.x;
    if (e >= E) return;
    int r = row[e];
    int pos = atomicAdd(&cnt[r], 1);
    if (pos < CAP) bucket[(size_t)r * CAP + pos] = e;
}

// ---- kernel 3: per node, select 10 smallest edge ids -> neighbor cols -----
__global__ void ps_select(const int* __restrict__ cnt,
                          const int* __restrict__ bucket,
                          const int* __restrict__ col,   // edge_index row 1
                          int* __restrict__ nbr,
                          int* __restrict__ kcnt,
                          int N) {
    int node = blockIdx.x * blockDim.x + threadIdx.x;
    if (node >= N) return;
    int deg = cnt[node];
    int m   = deg < CAP ? deg : CAP;
    int ids[CAP];
    const int* bk = bucket + (size_t)node * CAP;
    for (int i = 0; i < m; ++i) ids[i] = bk[i];
    int kk = deg < KNBR ? deg : KNBR;
    unsigned long long used = 0ull;
    for (int s = 0; s < kk; ++s) {
        int best = 0x7fffffff, bi = 0;
        for (int i = 0; i < m; ++i) {
            if (!((used >> i) & 1ull) && ids[i] < best) { best = ids[i]; bi = i; }
        }
        used |= (1ull << bi);
        nbr[(size_t)node * KNBR + s] = col[best];
    }
    kcnt[node] = kk;
}

// ---- kernel 4: fused aggregate + linear (f32 WMMA) ------------------------
// blockDim = 256 (8 waves).  Each block grid-strides over 16-node tiles.
// Phase A: 16 threads per node gather & mean neighbor rows into LDS tile.
// Phase B: wave w computes out[:, 16w:16w+16] = agg(16x128) @ W^T + b via
//          32 x v_wmma_f32_16x16x4_f32, with B fragments held in registers.
__global__ void ps_agg_linear(const float* __restrict__ x,
                              const float* __restrict__ W,   // [128,128] row-major
                              const float* __restrict__ b,
                              const int*   __restrict__ nbr,
                              const int*   __restrict__ kcnt,
                              float* __restrict__ out,
                              int N) {
    extern __shared__ float smem[];
    float* sWt  = smem;                       // [128][WT_LD]  W^T: sWt[k][n] = W[n][k]
    float* sAgg = sWt + D_FEAT * WT_LD;       // [16][AGG_LD]
    float* sB   = sAgg + 16 * AGG_LD;         // [128]

    const int tid = threadIdx.x;

    // stage W^T (coalesced global read, transposed LDS write) and bias
    for (int i = tid; i < D_FEAT * D_FEAT; i += blockDim.x) {
        int n = i >> 7;          // row of W
        int k = i & (D_FEAT - 1);
        sWt[k * WT_LD + n] = W[i];
    }
    for (int i = tid; i < D_FEAT; i += blockDim.x) sB[i] = b[i];
    __syncthreads();

    const int lane = tid & 31;
    const int wave = tid >> 5;           // 0..7 -> output col tile n0 = 16*wave
    const int n0   = wave << 4;
    const int lm   = lane & 15;
    const int kh   = (lane >> 4) << 1;   // 0 for lanes 0-15, 2 for lanes 16-31

    // ---- hoist B fragments (invariant across tiles) into registers --------
    // B (4x16) tile at k0: B[k][n] = W[n][k] = sWt[k][n]
    v2f bfrag[NKSTEP];
#pragma unroll
    for (int t = 0; t < NKSTEP; ++t) {
        int kk = t << 2;
        bfrag[t].x = sWt[(kk + kh)     * WT_LD + n0 + lm];
        bfrag[t].y = sWt[(kk + kh + 1) * WT_LD + n0 + lm];
    }
    const float bv = sB[n0 + lm];

    const int numTiles = (N + 15) >> 4;
    for (int tile = blockIdx.x; tile < numTiles; tile += gridDim.x) {
        const int node0 = tile << 4;

        // ---------------- Phase A: aggregation into sAgg -------------------
        {
            const int j  = tid >> 4;     // node within tile (0..15)
            const int tl = tid & 15;     // 16 threads cover a 128-f32 row
            const int node = node0 + j;
            float acc[8];
#pragma unroll
            for (int q = 0; q < 8; ++q) acc[q] = 0.0f;
            if (node < N) {
                int k = kcnt[node];
                if (k == 0) {
                    const float4* xr = (const float4*)(x + (size_t)node * D_FEAT);
                    float4 v0 = xr[tl * 2], v1 = xr[tl * 2 + 1];
                    acc[0] = v0.x; acc[1] = v0.y; acc[2] = v0.z; acc[3] = v0.w;
                    acc[4] = v1.x; acc[5] = v1.y; acc[6] = v1.z; acc[7] = v1.w;
                } else {
                    for (int i = 0; i < k; ++i) {
                        int c = nbr[(size_t)node * KNBR + i];
                        const float4* xr = (const float4*)(x + (size_t)c * D_FEAT);
                        float4 v0 = xr[tl * 2], v1 = xr[tl * 2 + 1];
                        acc[0] += v0.x; acc[1] += v0.y; acc[2] += v0.z; acc[3] += v0.w;
                        acc[4] += v1.x; acc[5] += v1.y; acc[6] += v1.z; acc[7] += v1.w;
                    }
                    float inv = 1.0f / (float)k;
#pragma unroll
                    for (int q = 0; q < 8; ++q) acc[q] *= inv;
                }
            }
            float* rowp = sAgg + j * AGG_LD + tl * 8;
#pragma unroll
            for (int q = 0; q < 8; ++q) rowp[q] = acc[q];
        }
        __syncthreads();

        // ---------------- Phase B: 16x128 @ 128x16 via f32 WMMA ------------
        v8f dacc;
#pragma unroll
        for (int v = 0; v < 8; ++v) dacc[v] = bv;

        const float* aRow = sAgg + lm * AGG_LD + kh;   // 8B-aligned (even float idx)
#pragma unroll
        for (int t = 0; t < NKSTEP; ++t) {
            // A (16x4): lanes 0-15 hold (K=kk,kk+1) of row lm; lanes 16-31 (kk+2,kk+3)
            v2f afr = *(const v2f*)(aRow + (t << 2));   // single ds_load_b64
            dacc = __builtin_amdgcn_wmma_f32_16x16x4_f32(
                false, afr, false, bfrag[t], (short)0, dacc, false, false);
        }

        // D layout: lane<16 -> m = v, lane>=16 -> m = v + 8; n = lane&15
        const int mbase = (lane >> 4) << 3;
#pragma unroll
        for (int v = 0; v < 8; ++v) {
            int m = mbase + v;
            int nodeM = node0 + m;
            if (nodeM < N) out[(size_t)nodeM * D_FEAT + n0 + lm] = dacc[v];
        }
        __syncthreads();   // sAgg reused next tile
    }
}

// ---------------------------------------------------------------------------
extern "C" void kernel_launch(void* const* d_in, const int* in_sizes, int n_in,
                              void* d_out, int out_size, void* d_ws, size_t ws_size,
                              hipStream_t stream) {
    const float* x  = (const float*)d_in[0];
    const int*   ei = (const int*)d_in[1];     // [2, E] row-major (int32)
    const float* W  = (const float*)d_in[2];   // [128, 128]
    const float* b  = (const float*)d_in[3];   // [128]
    float* out = (float*)d_out;

    const int N = in_sizes[0] / D_FEAT;
    const int E = in_sizes[1] / 2;

    // workspace layout
    int* cnt    = (int*)d_ws;                       // [N]
    int* bucket = cnt + N;                          // [N * CAP]
    int* nbr    = bucket + (size_t)N * CAP;         // [N * KNBR]
    int* kcnt   = nbr + (size_t)N * KNBR;           // [N]

    ps_zero_counts<<<(N + 255) / 256, 256, 0, stream>>>(cnt, N);
    ps_fill_buckets<<<(E + 255) / 256, 256, 0, stream>>>(ei, cnt, bucket, E);
    ps_select<<<(N + 255) / 256, 256, 0, stream>>>(cnt, bucket, ei + E, nbr, kcnt, N);

    size_t smemBytes = (size_t)(D_FEAT * WT_LD + 16 * AGG_LD + D_FEAT) * sizeof(float);
    ps_agg_linear<<<512, 256, smemBytes, stream>>>(x, W, b, nbr, kcnt, out, N);
}